// GCN_670014898212
// MI455X (gfx1250) — compile-verified
//
#include <hip/hip_runtime.h>

// ---------------------------------------------------------------------------
// GCN forward for MI455X (gfx1250, wave32).
// Fixed problem dims from the reference harness:
//   N=40000 nodes, E=640000 edges, D=128, M=256, G=2000 graphs.
// GEMMs use the CDNA5 fp32 matrix op V_WMMA_F32_16X16X4_F32 (full fp32
// precision; GEMM FLOPs are negligible vs the L2-resident edge scatter).
// ---------------------------------------------------------------------------

typedef float v2f __attribute__((ext_vector_type(2)));
typedef float v8f __attribute__((ext_vector_type(8)));

#define D_FEAT 128
#define M_HID  256
#define G_NUM  2000
#define BN_EPS 1e-5f

#if __has_builtin(__builtin_amdgcn_wmma_f32_16x16x4_f32)
#define USE_WMMA_F32 1
#else
#define USE_WMMA_F32 0
#endif

// ---------------- degree / norm ----------------
__global__ void k_deg_init(int* __restrict__ deg, int n) {
    int i = blockIdx.x * 256 + threadIdx.x;
    if (i < n) deg[i] = 1;                      // self-loop
}
__global__ void k_deg_count(const int* __restrict__ dst, int* __restrict__ deg, int e) {
    int i = blockIdx.x * 256 + threadIdx.x;
    if (i < e) atomicAdd(&deg[dst[i]], 1);
}
__global__ void k_dinv(const int* __restrict__ deg, float* __restrict__ dinv, int n) {
    int i = blockIdx.x * 256 + threadIdx.x;
    if (i < n) dinv[i] = rsqrtf((float)deg[i]); // deg >= 1 always
}

// ---------------- WMMA fp32 GEMM: C[M,N] = A[M,K] @ B[K,N] (+bias,relu) -----
// block = 128 threads (4 waves); wave w owns N-tile (blockIdx.y*4 + w).
// A tile (16 x K) staged in LDS, padded stride to avoid bank conflicts.
__global__ __launch_bounds__(128) void k_gemm(const float* __restrict__ A,
                                              const float* __restrict__ B,
                                              const float* __restrict__ bias,
                                              float* __restrict__ C,
                                              int M, int N, int K, int relu) {
    __shared__ float As[16 * (256 + 4)];
    const int kpad  = K + 4;
    const int mtile = blockIdx.x;
    const int tid   = threadIdx.x;

    for (int i = tid; i < 16 * K; i += 128) {
        int r = i / K, c = i - r * K;
        As[r * kpad + c] = A[(size_t)(mtile * 16 + r) * K + c];
    }
    __syncthreads();

    const int wave  = tid >> 5;
    const int lane  = tid & 31;
    const int ntile = blockIdx.y * 4 + wave;
    const int n     = lane & 15;
    const int ncol  = ntile * 16 + n;
    const int kh    = (lane >> 4) << 1;   // 0 or 2 (A/B fragment K-halves)
    const int m     = lane & 15;          // A fragment row
    const int mbase = (lane >> 4) << 3;   // 0 or 8 (C/D row base)

    v8f acc = {0.f, 0.f, 0.f, 0.f, 0.f, 0.f, 0.f, 0.f};

#if USE_WMMA_F32
    for (int k0 = 0; k0 < K; k0 += 4) {
        v2f a, b;
        a[0] = As[m * kpad + k0 + kh];
        a[1] = As[m * kpad + k0 + kh + 1];
        b[0] = B[(size_t)(k0 + kh) * N + ncol];
        b[1] = B[(size_t)(k0 + kh + 1) * N + ncol];
        acc = __builtin_amdgcn_wmma_f32_16x16x4_f32(false, a, false, b,
                                                    (short)0, acc, false, false);
    }
#else
    for (int k = 0; k < K; ++k) {
        float bv = B[(size_t)k * N + ncol];
#pragma unroll
        for (int r = 0; r < 8; ++r)
            acc[r] += As[(mbase + r) * kpad + k] * bv;
    }
#endif

    float bv = bias ? bias[ncol] : 0.f;
#pragma unroll
    for (int r = 0; r < 8; ++r) {
        float v = acc[r] + bv;
        if (relu) v = fmaxf(v, 0.f);
        C[(size_t)(mtile * 16 + mbase + r) * N + ncol] = v;
    }
}

// ---------------- aggregation ----------------
// agg[i,f] = bias[f] + h[i,f] * dinv[i]^2   (self-loop term folded in)
__global__ void k_agg_init(const float* __restrict__ h, const float* __restrict__ dinv,
                           const float* __restrict__ bias, float* __restrict__ agg,
                           int total) {
    int i = blockIdx.x * 256 + threadIdx.x;
    if (i >= total) return;
    int row = i >> 7, f = i & 127;
    float di = dinv[row];
    agg[i] = bias[f] + h[i] * di * di;
}
// 2 edges per 256-thread block; one feature per lane (512B coalesced rows).
__global__ void k_agg_edges(const float* __restrict__ h, const float* __restrict__ dinv,
                            const int* __restrict__ src, const int* __restrict__ dst,
                            float* __restrict__ agg, int e) {
    int edge = blockIdx.x * 2 + (threadIdx.x >> 7);
    if (edge >= e) return;
    int f = threadIdx.x & 127;
    int s = src[edge], d = dst[edge];
    float nrm = dinv[s] * dinv[d];
    atomicAdd(&agg[(size_t)d * D_FEAT + f], h[(size_t)s * D_FEAT + f] * nrm);
}

// ---------------- batchnorm ----------------
__global__ void k_bn_stats(const float* __restrict__ X, const float* __restrict__ g,
                           const float* __restrict__ b, float* __restrict__ scale,
                           float* __restrict__ shift, int nrows) {
    const int f = blockIdx.x;   // one block per feature column
    __shared__ float s_sum[256], s_sq[256];
    float sum = 0.f, sq = 0.f;
    for (int r = threadIdx.x; r < nrows; r += 256) {
        float v = X[(size_t)r * D_FEAT + f];
        sum += v; sq += v * v;
    }
    s_sum[threadIdx.x] = sum; s_sq[threadIdx.x] = sq;
    __syncthreads();
    for (int o = 128; o > 0; o >>= 1) {
        if (threadIdx.x < o) {
            s_sum[threadIdx.x] += s_sum[threadIdx.x + o];
            s_sq[threadIdx.x]  += s_sq[threadIdx.x + o];
        }
        __syncthreads();
    }
    if (threadIdx.x == 0) {
        float mu  = s_sum[0] / nrows;
        float var = s_sq[0] / nrows - mu * mu;
        float sc  = g[f] * rsqrtf(var + BN_EPS);
        scale[f] = sc;
        shift[f] = b[f] - mu * sc;
    }
}
__global__ void k_bn_relu(const float* __restrict__ x, const float* __restrict__ scale,
                          const float* __restrict__ shift, float* __restrict__ y,
                          int total) {
    int i = blockIdx.x * 256 + threadIdx.x;
    if (i >= total) return;
    int f = i & 127;
    y[i] = fmaxf(fmaf(x[i], scale[f], shift[f]), 0.f);
}

// ---------------- pooling ----------------
__global__ void k_zero(float* __restrict__ p, int n) {
    int i = blockIdx.x * 256 + threadIdx.x;
    if (i < n) p[i] = 0.f;
}
__global__ void k_pool(const float* __restrict__ h, const int* __restrict__ batch,
                       float* __restrict__ pooled) {
    int row = blockIdx.x, f = threadIdx.x;   // 128 threads
    atomicAdd(&pooled[(size_t)batch[row] * D_FEAT + f], h[(size_t)row * D_FEAT + f]);
}

// ---------------- final fc3: [G,256] @ [256,1] + b ----------------
__global__ void k_fc3(const float* __restrict__ X, const float* __restrict__ w,
                      const float* __restrict__ b, float* __restrict__ out) {
    int row  = blockIdx.x;
    int lane = threadIdx.x;   // 32 (wave32)
    float s = 0.f;
#pragma unroll
    for (int j = 0; j < M_HID; j += 32)
        s += X[(size_t)row * M_HID + j + lane] * w[j + lane];
    for (int o = 16; o > 0; o >>= 1) s += __shfl_xor(s, o, 32);
    if (lane == 0) out[row] = s + b[0];
}

// ---------------------------------------------------------------------------
extern "C" void kernel_launch(void* const* d_in, const int* in_sizes, int n_in,
                              void* d_out, int out_size, void* d_ws, size_t ws_size,
                              hipStream_t stream) {
    (void)n_in; (void)out_size; (void)ws_size;

    const float* x    = (const float*)d_in[0];
    const int*   ei   = (const int*)d_in[1];
    const int*   bat  = (const int*)d_in[2];
    const float* W1   = (const float*)d_in[4];
    const float* b1   = (const float*)d_in[5];
    const float* W2   = (const float*)d_in[6];
    const float* b2   = (const float*)d_in[7];
    const float* bn1g = (const float*)d_in[8],  *bn1b = (const float*)d_in[9];
    const float* bn2g = (const float*)d_in[10], *bn2b = (const float*)d_in[11];
    const float* fc1w = (const float*)d_in[12], *fc1b = (const float*)d_in[13];
    const float* fc2w = (const float*)d_in[14], *fc2b = (const float*)d_in[15];
    const float* fc3w = (const float*)d_in[16], *fc3b = (const float*)d_in[17];
    float* out = (float*)d_out;

    const int N = in_sizes[0] / D_FEAT;   // 40000
    const int E = in_sizes[1] / 2;        // 640000
    const int* src = ei;
    const int* dst = ei + E;

    // workspace carve-out (~67 MB total)
    char* ws = (char*)d_ws;
    auto carve = [&](size_t bytes) {
        char* p = ws; ws += (bytes + 255) & ~(size_t)255; return p;
    };
    int*   degc   = (int*)  carve((size_t)N * 4);
    float* dinv   = (float*)carve((size_t)N * 4);
    float* hA     = (float*)carve((size_t)N * D_FEAT * 4);
    float* hB     = (float*)carve((size_t)N * D_FEAT * 4);
    float* hC     = (float*)carve((size_t)N * D_FEAT * 4);
    float* sc     = (float*)carve(D_FEAT * 4);
    float* sh     = (float*)carve(D_FEAT * 4);
    float* pooled = (float*)carve((size_t)G_NUM * D_FEAT * 4);
    float* t1     = (float*)carve((size_t)G_NUM * M_HID * 4);
    float* t2     = (float*)carve((size_t)G_NUM * M_HID * 4);

    const int total = N * D_FEAT;
    const dim3 blk256(256);

    // degrees + rsqrt norm
    k_deg_init <<<(N + 255) / 256, blk256, 0, stream>>>(degc, N);
    k_deg_count<<<(E + 255) / 256, blk256, 0, stream>>>(dst, degc, E);
    k_dinv     <<<(N + 255) / 256, blk256, 0, stream>>>(degc, dinv, N);

    // layer 1: h = x @ W1 ; aggregate ; BN + relu
    k_gemm<<<dim3(N / 16, D_FEAT / 64), 128, 0, stream>>>(x, W1, nullptr, hA, N, D_FEAT, D_FEAT, 0);
    k_agg_init <<<(total + 255) / 256, blk256, 0, stream>>>(hA, dinv, b1, hB, total);
    k_agg_edges<<<(E + 1) / 2, blk256, 0, stream>>>(hA, dinv, src, dst, hB, E);
    k_bn_stats <<<D_FEAT, blk256, 0, stream>>>(hB, bn1g, bn1b, sc, sh, N);
    k_bn_relu  <<<(total + 255) / 256, blk256, 0, stream>>>(hB, sc, sh, hA, total);

    // layer 2
    k_gemm<<<dim3(N / 16, D_FEAT / 64), 128, 0, stream>>>(hA, W2, nullptr, hB, N, D_FEAT, D_FEAT, 0);
    k_agg_init <<<(total + 255) / 256, blk256, 0, stream>>>(hB, dinv, b2, hC, total);
    k_agg_edges<<<(E + 1) / 2, blk256, 0, stream>>>(hB, dinv, src, dst, hC, E);
    k_bn_stats <<<D_FEAT, blk256, 0, stream>>>(hC, bn2g, bn2b, sc, sh, N);
    k_bn_relu  <<<(total + 255) / 256, blk256, 0, stream>>>(hC, sc, sh, hA, total);

    // pooling: segment-sum nodes -> graphs
    k_zero<<<(G_NUM * D_FEAT + 255) / 256, blk256, 0, stream>>>(pooled, G_NUM * D_FEAT);
    k_pool<<<N, D_FEAT, 0, stream>>>(hA, bat, pooled);

    // MLP head
    k_gemm<<<dim3(G_NUM / 16, M_HID / 64), 128, 0, stream>>>(pooled, fc1w, fc1b, t1, G_NUM, M_HID, D_FEAT, 1);
    k_gemm<<<dim3(G_NUM / 16, M_HID / 64), 128, 0, stream>>>(t1, fc2w, fc2b, t2, G_NUM, M_HID, M_HID, 1);
    k_fc3 <<<G_NUM, 32, 0, stream>>>(t2, fc3w, fc3b, out);
}